// LSQ_quantizer_71631464563353
// MI455X (gfx1250) — compile-verified
//
#include <hip/hip_runtime.h>

// ============================================================================
// LSQ quantizer with random-sign Hadamard rotation, MI455X (gfx1250).
//
// hadamard = H_syl * diag(signs) / sqrt(2048)  (H_syl[0,:] == 1) =>
//   x @ H   = FWHT_syl(x) ⊙ H[0,:]
//   y @ H^T = FWHT_syl(y ⊙ H[0,:])
// Reduces 275 GFLOP of dense GEMM to ~1 GFLOP of butterflies: kernel becomes
// HBM-bound (~272 MB -> ~12 us at 23.3 TB/s). Dense-WMMA GEMM would need
// ~23 PFLOPS to match — above even dense-fp8 peak.
//
// One wave32 per row (2048 f32 resident in 64 VGPRs as 8 WMMA 16x16 tiles).
// Index bits j = [t2 t1 t0 | m3..m0 | n3..n0]; FWHT = one ± butterfly/bit:
//   Transform 1: n-levels via V_WMMA_F32_16X16X4_F32 (D = X * H16, A loaded
//     straight from global in A-layout, exact f32); m0..m2/t0..t2 in-register
//     VALU pairs; m3 via shfl_xor(16) with sign-xor trick (2 VALU/elem).
//   Transform 2: m-levels via WMMA (D = H16 * X, resident data marshaled into
//     the B operand with half-swaps); n-levels via shfl_xor(1,2,4,8) sign-xor
//     butterflies; t-levels VALU.
// Quantize: q = u*(h0/s); clamp(-Qn,Qp); v_rndne (half-to-even = jnp.round);
//   out = rint*(h0/s)*s^2  ( == rint*s*h0 ).
//
// x loads / out stores are non-temporal (single-use streams; keep the 192MB
// L2 for the wave-shared H[0,:] row instead). Row-loop control goes through
// readfirstlane so it is scalar: EXEC provably all-ones for every WMMA.
//
// WMMA f32 16x16x4 layouts (CDNA5 ISA):
//   A: M = lane&15, K = 2*(lane>>4)+reg                      [documented]
//   B: N = lane&15, K = 2*(lane>>4)+reg                      [assumed, mirrors
//      A; H16 is symmetric ±1 so the same regs serve as A and B operands]
//   C/D: N = lane&15, M = reg + 8*(lane>>4)                  [documented]
// ============================================================================

typedef float v2f __attribute__((ext_vector_type(2)));
typedef float v8f __attribute__((ext_vector_type(8)));

#define HAD_D 2048
#define TILES 8  // HAD_D / 256 elements per 16x16 tile

// Butterfly between VGPR slots v and v^(1<<bit) (m0..m2 levels).
__device__ __forceinline__ void bfly_reg(v8f (&T)[TILES], int bit) {
#pragma unroll
  for (int t = 0; t < TILES; ++t) {
#pragma unroll
    for (int v = 0; v < 8; ++v) {
      if ((v & (1 << bit)) == 0) {
        const int w = v | (1 << bit);
        const float a = T[t][v], b = T[t][w];
        T[t][v] = a + b;
        T[t][w] = a - b;
      }
    }
  }
}

// Butterfly between tiles t and t^(1<<bit) (t0..t2 levels).
__device__ __forceinline__ void bfly_tile(v8f (&T)[TILES], int bit) {
#pragma unroll
  for (int t = 0; t < TILES; ++t) {
    if ((t & (1 << bit)) == 0) {
      const int u = t | (1 << bit);
#pragma unroll
      for (int v = 0; v < 8; ++v) {
        const float a = T[t][v], b = T[u][v];
        T[t][v] = a + b;
        T[u][v] = a - b;
      }
    }
  }
}

// Cross-lane butterfly: new = (x ^ sign) + partner  -> v_xor + v_add (2 VALU).
__device__ __forceinline__ void bfly_lane(v8f (&T)[TILES], int mask, int lane) {
  const int sgn = (lane & mask) ? (int)0x80000000 : 0;
#pragma unroll
  for (int t = 0; t < TILES; ++t) {
#pragma unroll
    for (int v = 0; v < 8; ++v) {
      const float x = T[t][v];
      const float p = __shfl_xor(x, mask, 32);
      T[t][v] = __int_as_float(__float_as_int(x) ^ sgn) + p;
    }
  }
}

__global__ __launch_bounds__(256) void lsq_had_fwht_kernel(
    const float* __restrict__ x, const float* __restrict__ scale,
    const float* __restrict__ had, const int* __restrict__ Qn,
    const int* __restrict__ Qp, float* __restrict__ out, int rows) {
  const int lane = threadIdx.x & 31;
  const int lo = lane & 15;    // N index on lanes
  const int half = lane >> 4;  // lane-half

  // Forward value of s is just scale[0] (STE tricks only affect gradients).
  const float s = scale[0];
  const float inv_s = 1.0f / s;
  const float ss = s * s;
  const float qlo = -(float)Qn[0];
  const float qhi = (float)Qp[0];

  // Sylvester H16 chunk constants: HC[c][r] = H16[4c+2*half+r, lane&15].
  // Valid simultaneously as T1's B operand and T2's A operand.
  v2f HC[4];
#pragma unroll
  for (int c = 0; c < 4; ++c) {
#pragma unroll
    for (int r = 0; r < 2; ++r) {
      const int k = 4 * c + 2 * half + r;
      HC[c][r] = (__popc(k & lo) & 1) ? -1.0f : 1.0f;
    }
  }

  // Preload h0/s = H[0,:]/s in resident tile layout (L2-cached, wave-shared).
  v8f H0a[TILES];
#pragma unroll
  for (int t = 0; t < TILES; ++t) {
#pragma unroll
    for (int v = 0; v < 8; ++v) {
      H0a[t][v] = had[t * 256 + (v + 8 * half) * 16 + lo] * inv_s;
    }
  }

  // Scalar (wave-uniform) loop bookkeeping: EXEC stays all-ones for WMMA.
  const int wavesPerBlock = (int)(blockDim.x >> 5);
  const int waveInBlock = __builtin_amdgcn_readfirstlane((int)(threadIdx.x >> 5));
  const int wave = (int)blockIdx.x * wavesPerBlock + waveInBlock;
  const int nWaves = (int)gridDim.x * wavesPerBlock;

  for (int row = wave; row < rows; row += nWaves) {
    const float* __restrict__ xr = x + (size_t)row * HAD_D;
    v8f T[TILES];

    // ---- FWHT #1, n-levels: D_tile = X_tile x H16 (A streamed from global)
#pragma unroll
    for (int t = 0; t < TILES; ++t) {
      v8f acc = {};
#pragma unroll
      for (int c = 0; c < 4; ++c) {
        // A element (M=lo, K=2*half+r) at x[t*256 + lo*16 + 4c + 2*half + r]
        const v2f a = __builtin_nontemporal_load(
            (const v2f*)(xr + t * 256 + lo * 16 + 4 * c + 2 * half));
        acc = __builtin_amdgcn_wmma_f32_16x16x4_f32(
            false, a, false, HC[c], (short)0, acc, false, false);
      }
      T[t] = acc;
    }

    // ---- FWHT #1, m- and t-levels ----
    bfly_reg(T, 0);
    bfly_reg(T, 1);
    bfly_reg(T, 2);
    bfly_lane(T, 16, lane);
    bfly_tile(T, 0);
    bfly_tile(T, 1);
    bfly_tile(T, 2);

    // ---- quantize: y = rint(clip(u*h0/s,-Qn,Qp)) * s * h0 ----
#pragma unroll
    for (int t = 0; t < TILES; ++t) {
#pragma unroll
      for (int v = 0; v < 8; ++v) {
        const float ha = H0a[t][v];  // h0/s
        float q = T[t][v] * ha;
        q = fminf(fmaxf(q, qlo), qhi);
        T[t][v] = rintf(q) * ha * ss;  // rint * h0 * s
      }
    }

    // ---- FWHT #2, m-levels via WMMA: D_tile = H16 x X_tile ----
    // B operand row K = 4c + 2*half + r lives in T[t][slot] with
    // slot = 4*(c&1) + 2*half + r, source-half = c>>1; half-swap via
    // shfl_xor(16) when source-half != lane-half.
#pragma unroll
    for (int t = 0; t < TILES; ++t) {
      v8f acc = {};
#pragma unroll
      for (int c = 0; c < 4; ++c) {
        const int base = (c & 1) * 4;
        const bool upper = (c >> 1) != 0;
        v2f b;
#pragma unroll
        for (int r = 0; r < 2; ++r) {
          const float sA = T[t][base + r];      // rows m = base+{0,1} (+8*src)
          const float sB = T[t][base + 2 + r];  // rows m = base+{2,3} (+8*src)
          float val;
          if (!upper) {
            // sources in low halves: low lanes take sA, high take swapped sB
            val = half ? __shfl_xor(sB, 16, 32) : sA;
          } else {
            // sources in high halves: low lanes take swapped sA, high take sB
            val = half ? sB : __shfl_xor(sA, 16, 32);
          }
          b[r] = val;
        }
        acc = __builtin_amdgcn_wmma_f32_16x16x4_f32(
            false, HC[c], false, b, (short)0, acc, false, false);
      }
      T[t] = acc;  // all chunks of tile t consumed before overwrite
    }

    // ---- FWHT #2, n-levels (sign-xor shuffles) and t-levels ----
    bfly_lane(T, 1, lane);
    bfly_lane(T, 2, lane);
    bfly_lane(T, 4, lane);
    bfly_lane(T, 8, lane);
    bfly_tile(T, 0);
    bfly_tile(T, 1);
    bfly_tile(T, 2);

    // ---- store (resident layout -> linear row, non-temporal stream) ----
    float* __restrict__ o = out + (size_t)row * HAD_D;
#pragma unroll
    for (int t = 0; t < TILES; ++t) {
#pragma unroll
      for (int v = 0; v < 8; ++v) {
        __builtin_nontemporal_store(T[t][v],
                                    o + t * 256 + (v + 8 * half) * 16 + lo);
      }
    }
  }
}

extern "C" void kernel_launch(void* const* d_in, const int* in_sizes, int n_in,
                              void* d_out, int out_size, void* d_ws,
                              size_t ws_size, hipStream_t stream) {
  // setup_inputs order: x, scale, hadamard, Qn, Qp, num_elements
  const float* x = (const float*)d_in[0];
  const float* scale = (const float*)d_in[1];
  const float* had = (const float*)d_in[2];
  const int* Qn = (const int*)d_in[3];
  const int* Qp = (const int*)d_in[4];
  float* out = (float*)d_out;

  const int rows = in_sizes[0] / HAD_D;  // B*S = 16384
  const int wavesPerBlock = 8;           // 256 threads
  int blocks = (rows + wavesPerBlock - 1) / wavesPerBlock;
  if (blocks < 1) blocks = 1;

  lsq_had_fwht_kernel<<<blocks, 256, 0, stream>>>(x, scale, had, Qn, Qp, out,
                                                  rows);
}